// SparseNGCNLayer_36369783062753
// MI455X (gfx1250) — compile-verified
//
#include <hip/hip_runtime.h>

// ---------------------------------------------------------------------------
// SparseNGCNLayer for MI455X (gfx1250, wave32)
//   stage 1: sparse_feat @ W  + bias, relu  -> binned densify + f32 WMMA
//   stages 2-4: adj @ X  -> one wave32 per nnz, coalesced gather + f32 atomics
// ---------------------------------------------------------------------------

typedef float v2f __attribute__((ext_vector_type(2)));
typedef float v8f __attribute__((ext_vector_type(8)));

#define N_NODES   100000
#define IN_CH     512
#define OUT_CH    64
#define NBLK      (N_NODES / 16)     // 6250 row-blocks of 16
#define TILE_LDA  520                // padded LDS row stride (floats), breaks bank conflicts

// ---------------------------------------------------------------------------
__global__ void zero_f32_kernel(float* __restrict__ p, int n) {
    int t = blockIdx.x * blockDim.x + threadIdx.x;
    if (t < n) p[t] = 0.0f;
}

__global__ void zero_i32_kernel(int* __restrict__ p, int n) {
    int t = blockIdx.x * blockDim.x + threadIdx.x;
    if (t < n) p[t] = 0;
}

// Count feature nnz per 16-row block.
__global__ void count_kernel(const int* __restrict__ fidx, int nnz,
                             int* __restrict__ counts) {
    int t = blockIdx.x * blockDim.x + threadIdx.x;
    if (t < nnz) atomicAdd(&counts[fidx[t] >> 4], 1);
}

// Exclusive scan over NBLK block counts (single 256-thread workgroup).
__global__ void __launch_bounds__(256)
scan_kernel(const int* __restrict__ counts, int* __restrict__ offsets,
            int* __restrict__ cursors) {
    __shared__ int part[256];
    const int CH = (NBLK + 255) / 256;           // 25 bins per thread
    int tid = threadIdx.x;
    int s = 0;
    for (int j = 0; j < CH; ++j) {
        int i = tid * CH + j;
        if (i < NBLK) s += counts[i];
    }
    part[tid] = s;
    __syncthreads();
    // Hillis-Steele inclusive scan
    for (int off = 1; off < 256; off <<= 1) {
        int v = (tid >= off) ? part[tid - off] : 0;
        __syncthreads();
        part[tid] += v;
        __syncthreads();
    }
    int run = part[tid] - s;                     // exclusive base for this chunk
    for (int j = 0; j < CH; ++j) {
        int i = tid * CH + j;
        if (i < NBLK) {
            offsets[i] = run;
            cursors[i] = run;
            run += counts[i];
        }
    }
    if (tid == 255) offsets[NBLK] = part[255];
}

// Scatter nnz into per-block bins: packed = (row_in_block << 16) | col.
__global__ void scatter_kernel(const int* __restrict__ fidx,
                               const float* __restrict__ fval, int nnz,
                               int* __restrict__ cursors,
                               unsigned* __restrict__ packed,
                               float* __restrict__ pvals) {
    int t = blockIdx.x * blockDim.x + threadIdx.x;
    if (t >= nnz) return;
    int r = fidx[t];
    int c = fidx[nnz + t];
    int pos = atomicAdd(&cursors[r >> 4], 1);
    packed[pos] = ((unsigned)(r & 15) << 16) | (unsigned)c;
    pvals[pos]  = fval[t];
}

// Stage 1: per 16-row block, densify into LDS then dense GEMM with f32 WMMA.
// 128 threads = 4 waves; wave w owns output N-tile w (16 columns).
__global__ void __launch_bounds__(128)
stage1_wmma_kernel(const unsigned* __restrict__ packed,
                   const float* __restrict__ pvals,
                   const int* __restrict__ offsets,
                   const float* __restrict__ W,      // [IN_CH, OUT_CH] row-major
                   const float* __restrict__ bias,   // [OUT_CH]
                   float* __restrict__ X0) {         // [N_NODES, OUT_CH]
    __shared__ float tileA[16 * TILE_LDA];           // 16 x 512 (padded), 33 KB

    const int blk = blockIdx.x;
    const int tid = threadIdx.x;

    // zero dense tile
    for (int i = tid; i < 16 * TILE_LDA; i += 128) tileA[i] = 0.0f;
    __syncthreads();

    // scatter this block's nonzeros into the dense tile (LDS ds_add_f32)
    const int beg = offsets[blk], end = offsets[blk + 1];
    for (int i = beg + tid; i < end; i += 128) {
        unsigned p = packed[i];
        atomicAdd(&tileA[(p >> 16) * TILE_LDA + (p & 0xFFFFu)], pvals[i]);
    }
    __syncthreads();

    // dense 16x512 @ 512x64 via V_WMMA_F32_16X16X4_F32
    const int lane = tid & 31;
    const int wave = tid >> 5;          // N-tile index 0..3
    const int half = lane >> 4;         // 0: K+0/1, 1: K+2/3 (A/B operand layout)
    const int l16  = lane & 15;
    const int colB = wave * 16 + l16;   // output column for B operand / epilogue

    v8f acc = {};
    for (int k0 = 0; k0 < IN_CH; k0 += 4) {
        const int kk = k0 + half * 2;
        // A 16x4: lane<16 holds (M=l16, K=k0..k0+1), lane>=16 holds K=k0+2..k0+3
        v2f a = *(const v2f*)&tileA[l16 * TILE_LDA + kk];
        // B 4x16: mirrored layout, rows kk, kk+1 at column colB
        v2f b;
        b.x = W[kk * OUT_CH + colB];
        b.y = W[(kk + 1) * OUT_CH + colB];
        acc = __builtin_amdgcn_wmma_f32_16x16x4_f32(
            /*neg_a=*/false, a, /*neg_b=*/false, b,
            /*c_mod=*/(short)0, acc, /*reuse_a=*/false, /*reuse_b=*/false);
    }

    // epilogue: C/D layout -> VGPR j: lanes 0-15 M=j, lanes 16-31 M=j+8
    const float bv = bias[colB];
#pragma unroll
    for (int j = 0; j < 8; ++j) {
        int m = j + half * 8;
        float v = acc[j] + bv;
        v = v > 0.0f ? v : 0.0f;                 // relu
        X0[(blk * 16 + m) * OUT_CH + colB] = v;
    }
}

// Adjacency SpMM: one wave32 per nonzero; each lane handles 2 channels.
// Gathered row (256B) is L2-resident (dense operand is 25.6MB << 192MB L2).
__global__ void __launch_bounds__(256)
adj_spmm_kernel(const int* __restrict__ aidx, const float* __restrict__ avals,
                int nnz, const float* __restrict__ Xin, float* __restrict__ Xout) {
    int e = blockIdx.x * 8 + (threadIdx.x >> 5);
    if (e >= nnz) return;
    int lane = threadIdx.x & 31;
    int r = aidx[e];
    int c = aidx[nnz + e];
    float v = avals[e];
    const float2 x = *(const float2*)&Xin[c * OUT_CH + lane * 2];
    atomicAdd(&Xout[r * OUT_CH + lane * 2 + 0], v * x.x);
    atomicAdd(&Xout[r * OUT_CH + lane * 2 + 1], v * x.y);
}

// ---------------------------------------------------------------------------
extern "C" void kernel_launch(void* const* d_in, const int* in_sizes, int n_in,
                              void* d_out, int out_size, void* d_ws, size_t ws_size,
                              hipStream_t stream) {
    (void)n_in; (void)out_size; (void)ws_size;

    const int*   fidx = (const int*)d_in[0];     // [2, FEAT_NNZ]
    const float* fval = (const float*)d_in[1];   // [FEAT_NNZ]
    const int*   aidx = (const int*)d_in[2];     // [2, ADJ_NNZ]
    const float* aval = (const float*)d_in[3];   // [ADJ_NNZ]
    const float* W    = (const float*)d_in[4];   // [512, 64]
    const float* bias = (const float*)d_in[5];   // [1, 64]
    float*       out  = (float*)d_out;           // [N, 64]

    const int fnnz = in_sizes[1];
    const int annz = in_sizes[3];

    // workspace carve-out (256B aligned sections)
    char*  ws  = (char*)d_ws;
    size_t off = 0;
    auto carve = [&](size_t bytes) -> char* {
        char* p = ws + off;
        off = (off + bytes + 255) & ~(size_t)255;
        return p;
    };
    int*      counts  = (int*)carve((size_t)NBLK * 4);
    int*      cursors = (int*)carve((size_t)NBLK * 4);
    int*      offsets = (int*)carve((size_t)(NBLK + 1) * 4);
    unsigned* packed  = (unsigned*)carve((size_t)fnnz * 4);
    float*    pvals   = (float*)carve((size_t)fnnz * 4);
    float*    X0      = (float*)carve((size_t)N_NODES * OUT_CH * 4);
    float*    X1      = (float*)carve((size_t)N_NODES * OUT_CH * 4);

    const int nfeat = N_NODES * OUT_CH;

    // --- bin feature nnz by 16-row block ---
    zero_i32_kernel<<<(NBLK + 255) / 256, 256, 0, stream>>>(counts, NBLK);
    count_kernel<<<(fnnz + 255) / 256, 256, 0, stream>>>(fidx, fnnz, counts);
    scan_kernel<<<1, 256, 0, stream>>>(counts, offsets, cursors);
    scatter_kernel<<<(fnnz + 255) / 256, 256, 0, stream>>>(fidx, fval, fnnz,
                                                           cursors, packed, pvals);

    // --- stage 1: densify + WMMA GEMM (+bias, relu) -> X0 ---
    stage1_wmma_kernel<<<NBLK, 128, 0, stream>>>(packed, pvals, offsets, W, bias, X0);

    // --- stages 2-4: X1 = A@X0 ; X0 = A@X1 ; out = A@X0 ---
    zero_f32_kernel<<<(nfeat + 255) / 256, 256, 0, stream>>>(X1, nfeat);
    adj_spmm_kernel<<<(annz + 7) / 8, 256, 0, stream>>>(aidx, aval, annz, X0, X1);

    zero_f32_kernel<<<(nfeat + 255) / 256, 256, 0, stream>>>(X0, nfeat);
    adj_spmm_kernel<<<(annz + 7) / 8, 256, 0, stream>>>(aidx, aval, annz, X1, X0);

    zero_f32_kernel<<<(nfeat + 255) / 256, 256, 0, stream>>>(out, nfeat);
    adj_spmm_kernel<<<(annz + 7) / 8, 256, 0, stream>>>(aidx, aval, annz, X0, out);
}